// Model_18098992185387
// MI455X (gfx1250) — compile-verified
//
#include <hip/hip_runtime.h>
#include <hip/hip_bf16.h>
#include <math.h>

// GRU decode step on MI455X (gfx1250), wave32, single persistent WGP.
// Matvecs run on V_WMMA_F32_16X16X4_F32 (fp32-exact matrix pipe).

#define HID    139
#define HID3   417            // 3*HID
#define VOCAB  100
#define LAYERS 6
#define GCHUNK 27             // ceil(417/16)
#define DCHUNK 7              // ceil(100/16)
#define XPAD   144            // HID padded, x[139..143] = 0

typedef __attribute__((ext_vector_type(2))) float v2f;
typedef __attribute__((ext_vector_type(8))) float v8f;

__device__ __forceinline__ float sigmoidf_(float x) {
    return 1.0f / (1.0f + expf(-x));
}

// Computes out[j] = bias[j] + sum_k W[j][k]*v[k] for j in [chunk*16, chunk*16+16) ∩ [0,nrows)
// W: row-major [nrows x 139]. v: padded vector (>=140 entries, v[139]==0).
// A tile = 16 weight rows x 4 K; B tile = v chunk replicated across 16 columns,
// so every column of D holds the 16 dot products.
__device__ __forceinline__ void matvec16(const float* __restrict__ W,
                                         const float* __restrict__ v,
                                         const float* __restrict__ bias,
                                         float* __restrict__ out,
                                         int chunk, int nrows, int lane)
{
    int m = chunk * 16 + (lane & 15);
    if (m > nrows - 1) m = nrows - 1;          // clamp: keeps all loads in-bounds
    const float* row = W + (long)m * HID;
    const int off = (lane < 16) ? 0 : 2;       // ISA A/B layout: low lanes K={0,1}, high lanes K={2,3}

    v8f c = {};
    #pragma unroll 4
    for (int k0 = 0; k0 < 136; k0 += 4) {
        v2f a, b;
        a.x = row[k0 + off];
        a.y = row[k0 + off + 1];
        b = *(const v2f*)(v + k0 + off);       // LDS, 8B aligned (base 16B aligned, index even)
        c = __builtin_amdgcn_wmma_f32_16x16x4_f32(
                /*neg_a=*/false, a, /*neg_b=*/false, b,
                /*c_mod=*/(short)0, c, /*reuse_a=*/false, /*reuse_b=*/false);
    }
    {   // tail chunk: K = 136..139, element 139 does not exist in W -> mask A, v[139]==0 pads B
        v2f a, b;
        a.x = row[136 + off];                  // 136 or 138: valid
        float a1 = row[137];                   // valid for every lane
        a.y = (lane < 16) ? a1 : 0.0f;         // high lanes' K=3 slot (index 139) zeroed
        b = *(const v2f*)(v + 136 + off);      // v[139] == 0 by padding
        c = __builtin_amdgcn_wmma_f32_16x16x4_f32(
                false, a, false, b, (short)0, c, false, false);
    }
    // All columns of D are equal: lane 0 holds y[0..7] (M=0..7,N=0), lane 16 holds y[8..15]
    if ((lane & 15) == 0) {
        int base = chunk * 16 + ((lane >> 4) << 3);
        #pragma unroll
        for (int i = 0; i < 8; ++i) {
            int j = base + i;
            if (j < nrows) out[j] = c[i] + bias[j];
        }
    }
}

__launch_bounds__(864, 1)
__global__ void gru_decode_kernel(const int*   __restrict__ inp,
                                  const float* __restrict__ hidden,
                                  const float* __restrict__ emb,
                                  const float* __restrict__ w_ih,
                                  const float* __restrict__ w_hh,
                                  const float* __restrict__ b_ih,
                                  const float* __restrict__ b_hh,
                                  const float* __restrict__ dec_w,
                                  const float* __restrict__ dec_b,
                                  float*       __restrict__ out)
{
    __shared__ __align__(16) float x_lds[XPAD];     // current layer input (padded)
    __shared__ __align__(16) float h_lds[XPAD];     // hidden state of current layer (padded)
    __shared__ float gi_lds[HID3];
    __shared__ float gh_lds[HID3];

    const int tid    = threadIdx.x;
    const int lane   = tid & 31;
    const int wave   = tid >> 5;
    const int nwaves = (int)(blockDim.x >> 5);      // 27 waves

    // x = emb[token]; zero the pads once (pads are never overwritten afterwards)
    if (tid < XPAD) {
        int tok = inp[0];
        x_lds[tid] = (tid < HID) ? emb[tok * HID + tid] : 0.0f;
        h_lds[tid] = 0.0f;
    }
    __syncthreads();

    for (int l = 0; l < LAYERS; ++l) {
        if (tid < HID) h_lds[tid] = hidden[l * HID + tid];
        __syncthreads();

        const float* Wi = w_ih + (long)l * HID3 * HID;
        const float* Wh = w_hh + (long)l * HID3 * HID;
        const float* Bi = b_ih + l * HID3;
        const float* Bh = b_hh + l * HID3;

        // 54 wave-level WMMA jobs: 27 gi chunks + 27 gh chunks
        for (int cjob = wave; cjob < 2 * GCHUNK; cjob += nwaves) {
            if (cjob < GCHUNK)
                matvec16(Wi, x_lds, Bi, gi_lds, cjob, HID3, lane);
            else
                matvec16(Wh, h_lds, Bh, gh_lds, cjob - GCHUNK, HID3, lane);
        }
        __syncthreads();

        // gate math (PyTorch order r,z,n), writes new h into x_lds and the output
        if (tid < HID) {
            float hprev = h_lds[tid];
            float r = sigmoidf_(gi_lds[tid]            + gh_lds[tid]);
            float z = sigmoidf_(gi_lds[HID  + tid]     + gh_lds[HID  + tid]);
            float n = tanhf    (gi_lds[2*HID + tid] + r * gh_lds[2*HID + tid]);
            float h = (1.0f - z) * n + z * hprev;
            x_lds[tid] = h;
            out[VOCAB + l * HID + tid] = h;            // hidden_new, flat after logits
        }
        __syncthreads();
    }

    // decoder: logits = dec_w @ x + dec_b  (written straight to out[0..99])
    for (int cjob = wave; cjob < DCHUNK; cjob += nwaves) {
        matvec16(dec_w, x_lds, dec_b, out, cjob, VOCAB, lane);
    }
}

extern "C" void kernel_launch(void* const* d_in, const int* in_sizes, int n_in,
                              void* d_out, int out_size, void* d_ws, size_t ws_size,
                              hipStream_t stream) {
    const int*   inp    = (const int*)  d_in[0];
    const float* hidden = (const float*)d_in[1];
    const float* emb    = (const float*)d_in[2];
    const float* w_ih   = (const float*)d_in[3];
    const float* w_hh   = (const float*)d_in[4];
    const float* b_ih   = (const float*)d_in[5];
    const float* b_hh   = (const float*)d_in[6];
    const float* dec_w  = (const float*)d_in[7];
    const float* dec_b  = (const float*)d_in[8];
    float* out = (float*)d_out;

    hipLaunchKernelGGL(gru_decode_kernel, dim3(1), dim3(864), 0, stream,
                       inp, hidden, emb, w_ih, w_hh, b_ih, b_hh, dec_w, dec_b, out);
}